// GCNencoder_64364379898081
// MI455X (gfx1250) — compile-verified
//
#include <hip/hip_runtime.h>

typedef __attribute__((ext_vector_type(2))) float v2f;
typedef __attribute__((ext_vector_type(8))) float v8f;

#define NEG_SLOPE 0.01f

// ---------------------------------------------------------------------------
// Degree / normalization kernels
// ---------------------------------------------------------------------------
__global__ void k_init_deg(float* __restrict__ deg, int n) {
  int i = blockIdx.x * blockDim.x + threadIdx.x;
  if (i < n) deg[i] = 1.0f;  // self-loop contributes 1 to every node's degree
}

__global__ void k_accum_deg(const int* __restrict__ dst, float* __restrict__ deg, int E) {
  int e = blockIdx.x * blockDim.x + threadIdx.x;
  if (e < E) unsafeAtomicAdd(&deg[dst[e]], 1.0f);  // hw global_atomic_add_f32
}

__global__ void k_make_dinv(float* __restrict__ deg, int n) {
  int i = blockIdx.x * blockDim.x + threadIdx.x;
  if (i < n) deg[i] = rsqrtf(deg[i]);  // deg >= 1 always (self-loops)
}

// Per-edge norm coefficient: norm[e] = dinv[src]*dinv[dst]; self-loops at e>=E.
__global__ void k_edge_norm(const int* __restrict__ src, const int* __restrict__ dst,
                            const float* __restrict__ dinv, float* __restrict__ norm,
                            int E, int N) {
  int e = blockIdx.x * blockDim.x + threadIdx.x;
  if (e < E + N) {
    int s, d;
    if (e < E) { s = src[e]; d = dst[e]; }
    else       { s = d = e - E; }
    norm[e] = dinv[s] * dinv[d];
  }
}

__global__ void k_zero(float* __restrict__ p, long long n) {
  long long i = blockIdx.x * (long long)blockDim.x + threadIdx.x;
  if (i < n) p[i] = 0.0f;
}

// ---------------------------------------------------------------------------
// Dense GEMM via V_WMMA_F32_16X16X4_F32 (full fp32 precision).
// One wave computes a 16-row strip x NC columns.
//
// W is staged to LDS once per block, PRE-SWIZZLED into per-fragment order:
//   sW[(kstep*(NC/16) + t)*64 + lane*2 + j] = W[(kstep*4 + (lane>>4)*2 + j)*NC
//                                               + t*16 + (lane&15)]
// so each lane's WMMA B operand {W[ka,n], W[ka+1,n]} is one contiguous,
// 8B-aligned ds_load_b64 directly into the source VGPR pair (no repack movs).
//
// A frag: lanes 0-15 hold K=k0,k0+1; lanes 16-31 hold K=k0+2,k0+3 (one b64).
// C/D: VGPR r holds M=r (lanes 0-15) / M=r+8 (lanes 16-31), N=lane&15.
// EXEC stays all-ones through the WMMA loop; stores take an unguarded fast
// path when the whole strip is in range.
// ---------------------------------------------------------------------------
template <int K, int NC>
__global__ __launch_bounds__(128) void k_gemm_wmma(const float* __restrict__ X,
                                                   const float* __restrict__ W,
                                                   float* __restrict__ Out,
                                                   int nrows) {
  constexpr int NT = NC / 16;  // 16-wide column tiles
  __shared__ float sW[K * NC];

  // Stage + swizzle W into fragment order (cost paid once per block).
  for (int idx = threadIdx.x; idx < K * NC; idx += 128) {
    const int group  = idx >> 6;        // (kstep, tile)
    const int within = idx & 63;
    const int ln     = within >> 1;
    const int j      = within & 1;
    const int kstep  = group / NT;
    const int t      = group % NT;
    const int kk     = kstep * 4 + ((ln >> 4) << 1) + j;
    const int n      = t * 16 + (ln & 15);
    sW[idx] = W[kk * NC + n];
  }
  __syncthreads();

  const int lane = threadIdx.x & 31;
  const int wave = threadIdx.x >> 5;               // 4 waves / block
  const int row0 = (blockIdx.x * 4 + wave) * 16;   // 16-row strip per wave
  if (row0 >= nrows) return;                       // wave-uniform exit

  const int m  = lane & 15;
  const int hi = lane >> 4;                        // 0: lanes 0-15, 1: lanes 16-31
  int arow = row0 + m;
  if (arow >= nrows) arow = nrows - 1;             // clamp: keep EXEC all-ones

  const float* sWlane = sW + lane * 2;             // per-lane fragment base

  v8f acc[NT] = {};

  for (int k0 = 0; k0 < K; k0 += 4) {
    const int ka = k0 + hi * 2;                    // this lane's K pair
    v2f a = *(const v2f*)(X + (long long)arow * K + ka);
    const int kstep = k0 >> 2;
#pragma unroll
    for (int t = 0; t < NT; ++t) {
      v2f b = *(const v2f*)(sWlane + ((kstep * NT + t) << 6));
      acc[t] = __builtin_amdgcn_wmma_f32_16x16x4_f32(
          /*neg_a=*/false, a, /*neg_b=*/false, b,
          /*c_mod=*/(short)0, acc[t], /*reuse_a=*/false, /*reuse_b=*/false);
    }
  }

  if (row0 + 16 <= nrows) {
    // Fast path: whole strip valid -> straight-line stores, immediate offsets.
#pragma unroll
    for (int t = 0; t < NT; ++t) {
      float* op = Out + (long long)(row0 + hi * 8) * NC + t * 16 + m;
#pragma unroll
      for (int r = 0; r < 8; ++r) op[r * NC] = acc[t][r];
    }
  } else {
#pragma unroll
    for (int t = 0; t < NT; ++t) {
#pragma unroll
      for (int r = 0; r < 8; ++r) {
        const int rr = row0 + r + hi * 8;
        if (rr < nrows) Out[(long long)rr * NC + t * 16 + m] = acc[t][r];
      }
    }
  }
}

// ---------------------------------------------------------------------------
// Edge gather + scatter-add:  agg[dst] += H[src] * norm[e]
// Edges e in [0,E) from edge_idx; e in [E, E+N) are self-loops.
// float4 gather (H rows are L2-resident: <=12.8 MB << 192 MB L2),
// four hardware f32 atomics per thread.
// ---------------------------------------------------------------------------
template <int C>
__global__ void k_scatter(const float* __restrict__ H,
                          const int* __restrict__ src,
                          const int* __restrict__ dst,
                          const float* __restrict__ norm,
                          float* __restrict__ agg,
                          int E, int N) {
  const int LPE = C / 4;  // threads per edge
  long long gid = blockIdx.x * (long long)blockDim.x + threadIdx.x;
  long long e = gid / LPE;
  const int f = (int)(gid % LPE) * 4;
  const long long total = (long long)E + (long long)N;
  if (e >= total) return;
  int s, d;
  if (e < (long long)E) { s = src[e]; d = dst[e]; }
  else                  { s = d = (int)(e - E); }
  const float nrm = norm[e];
  const float4 hv = *(const float4*)(H + (long long)s * C + f);
  float* o = agg + (long long)d * C + f;
  unsafeAtomicAdd(o + 0, hv.x * nrm);
  unsafeAtomicAdd(o + 1, hv.y * nrm);
  unsafeAtomicAdd(o + 2, hv.z * nrm);
  unsafeAtomicAdd(o + 3, hv.w * nrm);
}

// ---------------------------------------------------------------------------
// Post-aggregation: bias + leaky_relu (+ dropout mask for layer 1)
// ---------------------------------------------------------------------------
__global__ void k_finalize1(float* __restrict__ agg, const float* __restrict__ b,
                            const float* __restrict__ mask, long long n) {
  long long i = blockIdx.x * (long long)blockDim.x + threadIdx.x;
  if (i < n) {
    float v = agg[i] + b[i & 63];
    v = v > 0.0f ? v : v * NEG_SLOPE;
    agg[i] = v * mask[i];
  }
}

__global__ void k_finalize2(float* __restrict__ out, const float* __restrict__ b,
                            long long n) {
  long long i = blockIdx.x * (long long)blockDim.x + threadIdx.x;
  if (i < n) {
    float v = out[i] + b[i & 31];
    out[i] = v > 0.0f ? v : v * NEG_SLOPE;
  }
}

// ---------------------------------------------------------------------------
// Launch
// ---------------------------------------------------------------------------
extern "C" void kernel_launch(void* const* d_in, const int* in_sizes, int n_in,
                              void* d_out, int out_size, void* d_ws, size_t ws_size,
                              hipStream_t stream) {
  const float* x    = (const float*)d_in[0];  // [N,64]
  const int*   eidx = (const int*)d_in[1];    // [2,E] flat: src then dst
  const float* mask = (const float*)d_in[2];  // [N,64]
  const float* W1   = (const float*)d_in[3];  // [64,64]
  const float* b1   = (const float*)d_in[4];  // [64]
  const float* W2   = (const float*)d_in[5];  // [64,32]
  const float* b2   = (const float*)d_in[6];  // [32]
  float* out = (float*)d_out;                 // [N,32]

  const int N = in_sizes[0] / 64;
  const int E = in_sizes[1] / 2;
  const int* src = eidx;
  const int* dst = eidx + E;

  // workspace layout (floats, regions rounded to 16B for float4 access):
  // dinv[N] | norm[E+N] | H1[N*64] | agg1[N*64] | H2[N*32]
  auto align4 = [](long long v) { return (v + 3) & ~3LL; };
  float* wsf  = (float*)d_ws;
  float* dinv = wsf;                 wsf += align4(N);
  float* norm = wsf;                 wsf += align4((long long)E + N);
  float* H1   = wsf;                 wsf += align4((long long)N * 64);
  float* agg1 = wsf;                 wsf += align4((long long)N * 64);
  float* H2   = wsf;

  const int B = 256;
  const int strips = (N + 15) / 16;
  const int gemmGrid = (strips + 3) / 4;

  // --- normalization coefficients ---
  k_init_deg<<<(N + B - 1) / B, B, 0, stream>>>(dinv, N);
  k_accum_deg<<<(E + B - 1) / B, B, 0, stream>>>(dst, dinv, E);
  k_make_dinv<<<(N + B - 1) / B, B, 0, stream>>>(dinv, N);
  k_edge_norm<<<(E + N + B - 1) / B, B, 0, stream>>>(src, dst, dinv, norm, E, N);

  // --- layer 1: H1 = x @ W1 ; agg1 = scatter(norm * H1[src]) ; act ---
  k_gemm_wmma<64, 64><<<gemmGrid, 128, 0, stream>>>(x, W1, H1, N);
  const long long n1 = (long long)N * 64;
  k_zero<<<(int)((n1 + B - 1) / B), B, 0, stream>>>(agg1, n1);
  const long long work1 = ((long long)E + N) * 16;
  k_scatter<64><<<(int)((work1 + B - 1) / B), B, 0, stream>>>(H1, src, dst, norm, agg1, E, N);
  k_finalize1<<<(int)((n1 + B - 1) / B), B, 0, stream>>>(agg1, b1, mask, n1);

  // --- layer 2: H2 = agg1 @ W2 ; d_out = scatter(norm * H2[src]) ; act ---
  k_gemm_wmma<64, 32><<<gemmGrid, 128, 0, stream>>>(agg1, W2, H2, N);
  const long long n2 = (long long)N * 32;
  k_zero<<<(int)((n2 + B - 1) / B), B, 0, stream>>>(out, n2);
  const long long work2 = ((long long)E + N) * 8;
  k_scatter<32><<<(int)((work2 + B - 1) / B), B, 0, stream>>>(H2, src, dst, norm, out, E, N);
  k_finalize2<<<(int)((n2 + B - 1) / B), B, 0, stream>>>(out, b2, n2);
}